// IF_7181185319482
// MI455X (gfx1250) — compile-verified
//
#include <hip/hip_runtime.h>
#include <stdint.h>

// ============================================================================
// IF-neuron (integrate, then T threshold/subtract steps) for MI455X (gfx1250).
//
// Bandwidth-bound: 252 MB in + 252 MB out -> ~22 us roofline at 23.3 TB/s.
// Strategy: wave32 streaming, 128-bit accesses, async Global->LDS ring
// (ASYNCcnt-tracked gfx1250 path) for the 63 strided slice loads, NT temporal
// hints on both load and store sides (504 MB dead stream vs 192 MB L2).
// ============================================================================

#define T_STEPS   63
#define BLOCK     256
#define RING      16   // LDS ring slots; 16 * 256 * 16B = 64 KB
#define IN_FLIGHT 8    // async loads in flight (< RING => 8-iter WAR gap)

typedef __attribute__((ext_vector_type(4))) float f32x4;

// gfx1250 async Global->LDS copy, 16B/lane, non-temporal (single-use stream).
// LDS dest byte offset in a VGPR; 64-bit global address in a VGPR pair.
// Tracked by ASYNCcnt.
#define ASYNC_LOAD_B128_NT(ldsaddr, gaddr)                                \
  asm volatile("global_load_async_to_lds_b128 %0, %1, off th:TH_LOAD_NT"  \
               :: "v"(ldsaddr), "v"(gaddr) : "memory")

#define WAIT_ASYNC_7() asm volatile("s_wait_asynccnt 7" ::: "memory")
#define WAIT_ASYNC_0() asm volatile("s_wait_asynccnt 0" ::: "memory")

__global__ __launch_bounds__(BLOCK) void if_spike_kernel(
    const f32x4* __restrict__ x,      // [T][n4] float4 slices
    const float* __restrict__ alpha_p,
    f32x4* __restrict__ out,          // [T][n4]
    int n4)                           // float4 elements per time slice
{
  __shared__ f32x4 stage[RING * BLOCK];   // 64 KB ring buffer

  const int    tid = threadIdx.x;
  const size_t p4  = (size_t)blockIdx.x * BLOCK + tid;
  const float  a   = *alpha_p;

  // Byte stride between consecutive time slices (4 MB here).
  const uint64_t slice_bytes = (uint64_t)n4 * sizeof(f32x4);

  // Wave-relative LDS byte offset of this lane's slot 0. Generic pointers to
  // LDS carry the offset in addr[31:0] (ISA 10.2 aperture mapping).
  const unsigned lds0      = (unsigned)(uintptr_t)(&stage[tid]);
  const unsigned slot_step = BLOCK * sizeof(f32x4);           // 4096 B
  const unsigned lds_last  = lds0 + (RING - 1) * slot_step;

  // ---- Phase 1 prologue: fill the async pipeline -------------------------
  uint64_t g_issue = (uint64_t)(uintptr_t)(x + p4);
#pragma unroll
  for (int i = 0; i < IN_FLIGHT; ++i) {
    ASYNC_LOAD_B128_NT(lds0 + (unsigned)i * slot_step, g_issue);
    g_issue += slice_bytes;
  }

  f32x4 m;
  m.x = m.y = m.z = m.w = 0.5f * a;

  // ---- Phase 1 steady state: wait oldest, consume, issue next ------------
  // Slot of the next *issue* rotates through the ring IN_FLIGHT ahead of the
  // consume slot, so an async LDS write never races the ds_load it replaces.
  unsigned lds_issue = lds0 + IN_FLIGHT * slot_step;
  for (int t = 0; t < T_STEPS - IN_FLIGHT; ++t) {
    WAIT_ASYNC_7();  // async loads complete in order => load t has landed
    f32x4 v = stage[(t & (RING - 1)) * BLOCK + tid];   // ds_load_b128
    m.x += v.x; m.y += v.y; m.z += v.z; m.w += v.w;

    ASYNC_LOAD_B128_NT(lds_issue, g_issue);
    g_issue  += slice_bytes;
    lds_issue = (lds_issue == lds_last) ? lds0 : lds_issue + slot_step;
  }

  // ---- Phase 1 tail: drain remaining IN_FLIGHT slots ---------------------
  WAIT_ASYNC_0();
#pragma unroll
  for (int t = T_STEPS - IN_FLIGHT; t < T_STEPS; ++t) {
    f32x4 v = stage[(t & (RING - 1)) * BLOCK + tid];
    m.x += v.x; m.y += v.y; m.z += v.z; m.w += v.w;
  }

  // ---- Phase 2: T threshold/subtract steps, streaming NT stores ----------
  // out[t] = alpha * spike ; m -= alpha * spike   (reset-by-subtraction)
  f32x4* o = out + p4;
  for (int t = 0; t < T_STEPS; ++t) {
    f32x4 s;
    s.x = (m.x > a) ? a : 0.0f;  m.x -= s.x;
    s.y = (m.y > a) ? a : 0.0f;  m.y -= s.y;
    s.z = (m.z > a) ? a : 0.0f;  m.z -= s.z;
    s.w = (m.w > a) ? a : 0.0f;  m.w -= s.w;
    __builtin_nontemporal_store(s, o + (size_t)t * n4);  // b128 NT store
  }
}

extern "C" void kernel_launch(void* const* d_in, const int* in_sizes, int n_in,
                              void* d_out, int out_size, void* d_ws, size_t ws_size,
                              hipStream_t stream) {
  (void)n_in; (void)out_size; (void)d_ws; (void)ws_size;
  const float* x     = (const float*)d_in[0];
  const float* alpha = (const float*)d_in[1];
  float*       out   = (float*)d_out;

  const int total = in_sizes[0];          // T * B * S * D = 66,060,288
  const int n     = total / T_STEPS;      // elements per time slice (1,048,576)
  const int n4    = n / 4;                // float4 per slice (262,144)

  dim3 grid(n4 / BLOCK), block(BLOCK);    // 1024 blocks x 256 threads (8 waves)
  if_spike_kernel<<<grid, block, 0, stream>>>(
      (const f32x4*)x, alpha, (f32x4*)out, n4);
}